// GraphStack_55808805044590
// MI455X (gfx1250) — compile-verified
//
#include <hip/hip_runtime.h>

typedef __attribute__((ext_vector_type(2))) float v2f;
typedef __attribute__((ext_vector_type(8))) float v8f;

#define GN_N 100000
#define GN_E 3200000LL
#define GN_DIN 128
#define GN_C 64
#define GN_EPS 1e-5f

// ---------------------------------------------------------------------------
// Utility: zero a float buffer
// ---------------------------------------------------------------------------
__global__ void gs_zero_f32(float* __restrict__ p, long long n) {
  long long t = (long long)blockIdx.x * blockDim.x + threadIdx.x;
  if (t < n) p[t] = 0.0f;
}

// ---------------------------------------------------------------------------
// Degree: deg[dst] += 1 per edge (fp32 atomics), then dinv = rsqrt(1+deg)
// ---------------------------------------------------------------------------
__global__ void gs_edge_degree(const long long* __restrict__ dst,
                               float* __restrict__ deg, long long E) {
  long long e = (long long)blockIdx.x * blockDim.x + threadIdx.x;
  if (e < E) atomicAdd(&deg[dst[e]], 1.0f);
}

__global__ void gs_finish_dinv(float* __restrict__ deg, int N) {
  int i = blockIdx.x * blockDim.x + threadIdx.x;
  if (i < N) deg[i] = rsqrtf(1.0f + deg[i]);
}

// ---------------------------------------------------------------------------
// WMMA f32 GEMM: out[M,64] = A[M,K] @ W[K,64].  M % 16 == 0, K % 4 == 0.
// One wave computes a 16x16 output tile via V_WMMA_F32_16X16X4_F32.
// 4 waves per block cover the 64-wide N dimension; blockIdx.x walks M tiles.
//
// Fragment layouts per CDNA5 ISA 7.12.2 (f32):
//   A 16x4 : lane l<16 holds {A[l][k0], A[l][k0+1]}; lanes 16-31 hold K+2.
//   B 4x16 : VGPR0 = rows {k0, k0+2} striped over lane halves, VGPR1 = +1.
//   C/D    : VGPR r = rows {m0+r, m0+r+8} striped over lane halves.
// ---------------------------------------------------------------------------
__global__ __launch_bounds__(128)
void gs_gemm_wmma_f32(const float* __restrict__ A, const float* __restrict__ W,
                      float* __restrict__ out, int K) {
  const int lane = threadIdx.x & 31;
  const int wv   = threadIdx.x >> 5;       // 0..3 -> N tile
  const int m0   = blockIdx.x * 16;
  const int n0   = wv * 16;
  const int mrow = lane & 15;
  const int ncol = lane & 15;
  const int koff = (lane >> 4) * 2;        // 0 for lanes 0-15, 2 for 16-31

  const float* __restrict__ arow = A + (size_t)(m0 + mrow) * K;

  v8f acc = {};
  for (int k0 = 0; k0 < K; k0 += 4) {
    v2f a, b;
    a.x = arow[k0 + koff];
    a.y = arow[k0 + koff + 1];
    b.x = W[(size_t)(k0 + koff)     * GN_C + n0 + ncol];
    b.y = W[(size_t)(k0 + koff + 1) * GN_C + n0 + ncol];
    acc = __builtin_amdgcn_wmma_f32_16x16x4_f32(
        /*neg_a=*/false, a, /*neg_b=*/false, b,
        /*c_mod=*/(short)0, acc, /*reuse_a=*/false, /*reuse_b=*/false);
  }

  const int rbase = m0 + ((lane >> 4) ? 8 : 0);
#pragma unroll
  for (int r = 0; r < 8; ++r)
    out[(size_t)(rbase + r) * GN_C + n0 + ncol] = acc[r];
}

// ---------------------------------------------------------------------------
// agg[n][c] = xw[n][c] * dinv[n]^2 + bias[c]   (self-loop term + bias)
// ---------------------------------------------------------------------------
__global__ void gs_init_agg(const float* __restrict__ xw,
                            const float* __restrict__ dinv,
                            const float* __restrict__ bias,
                            float* __restrict__ agg, long long total) {
  long long t = (long long)blockIdx.x * blockDim.x + threadIdx.x;
  if (t >= total) return;
  int n = (int)(t >> 6);
  int c = (int)(t & 63);
  float di = dinv[n];
  agg[t] = xw[t] * di * di + bias[c];
}

// ---------------------------------------------------------------------------
// Edge scatter: 16 threads per edge, each owning 4 channels (float4 gather,
// 4 f32 atomics). Consecutive threads cover one edge's 256-byte row ->
// fully coalesced gathers; tables are L2-resident.
// ---------------------------------------------------------------------------
__global__ void gs_edge_scatter(const long long* __restrict__ src,
                                const long long* __restrict__ dst,
                                const float* __restrict__ xw,
                                const float* __restrict__ dinv,
                                float* __restrict__ agg, long long E) {
  long long t = (long long)blockIdx.x * blockDim.x + threadIdx.x;
  long long total = E * 16;
  if (t >= total) return;
  long long e = t >> 4;
  int cg = (int)(t & 15) * 4;
  long long s = src[e];
  long long d = dst[e];
  float coef = dinv[s] * dinv[d];
  const float4 v = *(const float4*)(xw + s * GN_C + cg);
  float* p = agg + d * GN_C + cg;
  atomicAdd(p + 0, v.x * coef);
  atomicAdd(p + 1, v.y * coef);
  atomicAdd(p + 2, v.z * coef);
  atomicAdd(p + 3, v.w * coef);
}

// ---------------------------------------------------------------------------
// Per-channel sum & sum-of-squares: block covers 64 rows; LDS reduce across
// the 4 row-groups, then 128 atomics per block onto stats[0:64]=sum,
// stats[64:128]=sumsq.
// ---------------------------------------------------------------------------
__global__ __launch_bounds__(256)
void gs_stats_reduce(const float* __restrict__ agg,
                     float* __restrict__ stats, int N) {
  __shared__ float ssum[256];
  __shared__ float ssq[256];
  const int c    = threadIdx.x & 63;
  const int rgrp = threadIdx.x >> 6;   // 0..3
  const int base = blockIdx.x * 64;

  float sum = 0.0f, sq = 0.0f;
  for (int r = rgrp; r < 64; r += 4) {
    int row = base + r;
    if (row < N) {
      float v = agg[(size_t)row * GN_C + c];
      sum += v;
      sq  += v * v;
    }
  }
  ssum[threadIdx.x] = sum;
  ssq[threadIdx.x]  = sq;
  __syncthreads();
  if (threadIdx.x < 64) {
    sum = ssum[c] + ssum[c + 64] + ssum[c + 128] + ssum[c + 192];
    sq  = ssq[c]  + ssq[c + 64]  + ssq[c + 128]  + ssq[c + 192];
    atomicAdd(&stats[c], sum);
    atomicAdd(&stats[64 + c], sq);
  }
}

// ---------------------------------------------------------------------------
// Fold GraphNorm into per-channel scale/shift:
//   m   = E[x],  var = E[x^2] - 2*a*m^2 + a^2*m^2  (since xc = x - a*m)
//   y   = x * (gamma*rinv) + (beta - gamma*rinv*a*m)
// ---------------------------------------------------------------------------
__global__ void gs_norm_params(const float* __restrict__ stats,
                               const float* __restrict__ alpha,
                               const float* __restrict__ gamma,
                               const float* __restrict__ beta,
                               float* __restrict__ params, float invN) {
  int c = threadIdx.x;
  if (c >= 64) return;
  float m   = stats[c] * invN;
  float ex2 = stats[64 + c] * invN;
  float a   = alpha[c];
  float var = ex2 - 2.0f * a * m * m + a * a * m * m;
  float rinv = rsqrtf(var + GN_EPS);
  float s = gamma[c] * rinv;
  params[c]      = s;
  params[64 + c] = beta[c] - s * a * m;
}

__global__ void gs_apply_norm(const float* __restrict__ agg,
                              const float* __restrict__ params,
                              float* __restrict__ out, long long total) {
  long long t = (long long)blockIdx.x * blockDim.x + threadIdx.x;
  if (t >= total) return;
  int c = (int)(t & 63);
  out[t] = fmaf(agg[t], params[c], params[64 + c]);
}

// ---------------------------------------------------------------------------
// Host-side orchestration (graph-capture safe: stream launches only)
// ---------------------------------------------------------------------------
static inline size_t gs_align256(size_t x) { return (x + 255) & ~(size_t)255; }

extern "C" void kernel_launch(void* const* d_in, const int* in_sizes, int n_in,
                              void* d_out, int out_size, void* d_ws, size_t ws_size,
                              hipStream_t stream) {
  (void)in_sizes; (void)n_in; (void)out_size; (void)ws_size;

  const float*     x     = (const float*)d_in[0];          // [N,128]
  const long long* ei    = (const long long*)d_in[1];      // [2,E] int64
  const float*     W0    = (const float*)d_in[2];          // [128,64]
  const float*     Ws    = (const float*)d_in[3];          // [2,64,64]
  const float*     bias  = (const float*)d_in[4];          // [3,64]
  const float*     alpha = (const float*)d_in[5];          // [3,64]
  const float*     gamma = (const float*)d_in[6];          // [3,64]
  const float*     beta  = (const float*)d_in[7];          // [3,64]
  float*           out   = (float*)d_out;                  // [N,64]

  const long long* src = ei;
  const long long* dst = ei + GN_E;

  // Workspace layout (256B aligned slices)
  char* ws = (char*)d_ws;
  size_t off = 0;
  float* dinv = (float*)(ws + off);  off = gs_align256(off + (size_t)GN_N * 4);
  float* xw   = (float*)(ws + off);  off = gs_align256(off + (size_t)GN_N * GN_C * 4);
  float* agg  = (float*)(ws + off);  off = gs_align256(off + (size_t)GN_N * GN_C * 4);
  float* stats  = (float*)(ws + off); off = gs_align256(off + 128 * 4);
  float* params = (float*)(ws + off); off = gs_align256(off + 128 * 4);

  const long long NC = (long long)GN_N * GN_C;
  const int B = 256;

  // --- degree -> dinv ---
  gs_zero_f32<<<(int)((GN_N + B - 1) / B), B, 0, stream>>>(dinv, GN_N);
  gs_edge_degree<<<(int)((GN_E + B - 1) / B), B, 0, stream>>>(dst, dinv, GN_E);
  gs_finish_dinv<<<(int)((GN_N + B - 1) / B), B, 0, stream>>>(dinv, GN_N);

  // --- 3 GCN + GraphNorm layers ---
  for (int layer = 0; layer < 3; ++layer) {
    const float* H = (layer == 0) ? x : agg;
    const int    K = (layer == 0) ? GN_DIN : GN_C;
    const float* W = (layer == 0) ? W0 : (Ws + (size_t)(layer - 1) * GN_C * GN_C);

    // xw = H @ W   (WMMA f32 16x16x4; M=100000 is a multiple of 16)
    gs_gemm_wmma_f32<<<GN_N / 16, 128, 0, stream>>>(H, W, xw, K);

    // agg = xw * dinv^2 + bias
    gs_init_agg<<<(int)((NC + B - 1) / B), B, 0, stream>>>(
        xw, dinv, bias + layer * GN_C, agg, NC);

    // agg[dst] += xw[src] * dinv[src]*dinv[dst]
    long long ethreads = GN_E * 16;
    gs_edge_scatter<<<(int)((ethreads + B - 1) / B), B, 0, stream>>>(
        src, dst, xw, dinv, agg, GN_E);

    // GraphNorm: stats -> params -> fused scale/shift
    gs_zero_f32<<<1, 128, 0, stream>>>(stats, 128);
    gs_stats_reduce<<<(GN_N + 63) / 64, 256, 0, stream>>>(agg, stats, GN_N);
    gs_norm_params<<<1, 64, 0, stream>>>(
        stats, alpha + layer * GN_C, gamma + layer * GN_C, beta + layer * GN_C,
        params, 1.0f / (float)GN_N);

    float* dstbuf = (layer == 2) ? out : agg;   // in-place for layers 0,1
    gs_apply_norm<<<(int)((NC + B - 1) / B), B, 0, stream>>>(agg, params, dstbuf, NC);
  }
}